// ProposalLayer_3925600109282
// MI455X (gfx1250) — compile-verified
//
#include <hip/hip_runtime.h>

typedef __attribute__((ext_vector_type(2))) float v2f;
typedef __attribute__((ext_vector_type(8))) float v8f;

#define CIN   384
#define NY    200
#define NX    176
#define NB    4
#define PLANE (NY * NX)        /* 35200 */
#define NPIX  (NB * PLANE)     /* 140800 */
#define NCLS  6                /* cls output channels */
#define NOUT  48               /* 6 cls + 42 reg */
#define KSTEPS_PER_PHASE 48    /* K chunk of 192 */
#define FRAGS_PER_PHASE (KSTEPS_PER_PHASE * 3 * 32) /* 4608 float2 frags */

__global__ __launch_bounds__(256)
void proposal_head_kernel(const float* __restrict__ fm,
                          const float* __restrict__ Wc,
                          const float* __restrict__ bc,
                          const float* __restrict__ Wr,
                          const float* __restrict__ br,
                          float* __restrict__ out)
{
    // Weight fragments in per-lane WMMA B order: 36,864 bytes
    __shared__ float sW[FRAGS_PER_PHASE * 2];

    const int tid   = threadIdx.x;
    const int lane  = tid & 31;
    const int wv    = tid >> 5;
    const int mtile = blockIdx.x * 8 + wv;   // one 16-pixel tile per wave
    const int m0    = mtile * 16;
    const int b     = m0 / PLANE;            // tiles never straddle batches
    const int rem0  = m0 - b * PLANE;        // y*NX + x of first pixel

    const int nlo   = lane & 15;             // N column (B/C/D) or M row (A)
    const int khalf = (lane >> 4) * 2;       // K sub-offset within 4-step: 0 or 2

    // A element X[m0+nlo, k] lives at fm[b*CIN*PLANE + k*PLANE + rem0 + nlo]
    const float* aBase = fm + (size_t)b * CIN * PLANE + (size_t)(rem0 + nlo);

    v8f acc0 = {}, acc1 = {}, acc2 = {};

    for (int phase = 0; phase < 2; ++phase) {
        const int k0 = phase * (KSTEPS_PER_PHASE * 4);
        if (phase) __syncthreads();
        // Cooperatively stage weight fragments into LDS, pre-swizzled so the
        // inner loop is a single conflict-free ds_load_b64 per B fragment.
        for (int f = tid; f < FRAGS_PER_PHASE; f += 256) {
            const int l = f & 31;
            const int t = (f >> 5) % 3;
            const int s = f / 96;
            const int n = t * 16 + (l & 15);
            const int k = k0 + s * 4 + ((l >> 4) << 1);
            const float* src = (n < NCLS) ? (Wc + n * CIN + k)
                                          : (Wr + (n - NCLS) * CIN + k);
            sW[2 * f]     = src[0];
            sW[2 * f + 1] = src[1];
        }
        __syncthreads();

#pragma unroll 4
        for (int s = 0; s < KSTEPS_PER_PHASE; ++s) {
            const int kb = k0 + s * 4 + khalf;
            v2f a;
            a[0] = aBase[(size_t)kb * PLANE];        // coalesced 64B per half-wave
            a[1] = aBase[(size_t)(kb + 1) * PLANE];
            const float* wrow = &sW[(size_t)((s * 3) * 32 + lane) * 2];
            const v2f b0 = *(const v2f*)(wrow);
            const v2f b1 = *(const v2f*)(wrow + 64);
            const v2f b2 = *(const v2f*)(wrow + 128);
            acc0 = __builtin_amdgcn_wmma_f32_16x16x4_f32(
                false, a, false, b0, (short)0, acc0, false, false);
            acc1 = __builtin_amdgcn_wmma_f32_16x16x4_f32(
                false, a, false, b1, (short)0, acc1, false, false);
            acc2 = __builtin_amdgcn_wmma_f32_16x16x4_f32(
                false, a, false, b2, (short)0, acc2, false, false);
        }
    }

    // Epilogue: bias + scatter into the reference's output layouts.
    // Accumulator VGPR i: lanes 0-15 -> pixel m0+i, lanes 16-31 -> pixel m0+i+8.
    const int rowOff = (lane >> 4) * 8;
    float* outReg = out + (size_t)NCLS * NB * PLANE; // cls block = 844800 floats

#pragma unroll
    for (int t = 0; t < 3; ++t) {
        const v8f acc = (t == 0) ? acc0 : ((t == 1) ? acc1 : acc2);
        const int n = t * 16 + nlo;
        const float bias = (n < NCLS) ? bc[n] : br[n - NCLS];
        if (n < NCLS) {
            // cls_map (B,3,2,ny,nx): flat channel n -> contiguous 8-pixel run
            float* dst = out + ((size_t)(b * NCLS + n)) * PLANE + rem0 + rowOff;
#pragma unroll
            for (int i = 0; i < 8; ++i) dst[i] = acc[i] + bias;
        } else {
            // reg channel r=(cls*7+dof)*2+yaw -> (B,3,2,ny,nx,7) with DOF innermost
            const int r   = n - NCLS;
            const int c   = r / 14;
            const int w14 = r % 14;
            const int dof = w14 >> 1;
            const int yaw = w14 & 1;
            float* dst = outReg +
                ((size_t)((b * 3 + c) * 2 + yaw) * PLANE + (size_t)(rem0 + rowOff)) * 7 + dof;
#pragma unroll
            for (int i = 0; i < 8; ++i) dst[(size_t)i * 7] = acc[i] + bias;
        }
    }
}

extern "C" void kernel_launch(void* const* d_in, const int* in_sizes, int n_in,
                              void* d_out, int out_size, void* d_ws, size_t ws_size,
                              hipStream_t stream) {
    const float* fm = (const float*)d_in[0];
    const float* Wc = (const float*)d_in[1];
    const float* bc = (const float*)d_in[2];
    const float* Wr = (const float*)d_in[3];
    const float* br = (const float*)d_in[4];
    float* out = (float*)d_out;

    // 140800 pixels / (8 waves * 16 pixels) = 1100 blocks exactly
    dim3 grid(NPIX / 128);
    proposal_head_kernel<<<grid, dim3(256), 0, stream>>>(fm, Wc, bc, Wr, br, out);
}